// MGNN_3401614098765
// MI455X (gfx1250) — compile-verified
//
#include <hip/hip_runtime.h>

typedef float v2f __attribute__((ext_vector_type(2)));
typedef float v8f __attribute__((ext_vector_type(8)));

#define NNODES 100000
#define NEDGES 1600000

// ---------------- degree / normalization ----------------

__global__ void init_deg(float* deg, int n) {
    int i = blockIdx.x * blockDim.x + threadIdx.x;
    if (i < n) deg[i] = 1.0f;  // self-loop contributes 1 to every node's degree
}

__global__ void count_deg(const int* __restrict__ dst, float* deg, int e) {
    int i = blockIdx.x * blockDim.x + threadIdx.x;
    if (i < e) atomicAdd(&deg[dst[i]], 1.0f);
}

__global__ void deg_to_dinv(float* deg, int n) {
    int i = blockIdx.x * blockDim.x + threadIdx.x;
    if (i < n) deg[i] = rsqrtf(deg[i]);  // deg >= 1 always (self-loop)
}

__global__ void edge_norm(const int* __restrict__ src, const int* __restrict__ dst,
                          const float* __restrict__ dinv, float* __restrict__ norm, int e) {
    int i = blockIdx.x * blockDim.x + threadIdx.x;
    if (i < e) norm[i] = dinv[src[i]] * dinv[dst[i]];
}

// ---------------- fp32 WMMA GEMM: H = X @ W ----------------
// X: [N, KD] row-major, W: [KD, FOUT] row-major, Hout: [N, OST] row-major.
// Block = 128 threads = 4 waves; each wave owns one 16-column tile of a 16-row tile.
// N is a multiple of 16 (100000 = 6250*16) so no row guards.
template <int KD, int FOUT, int OST>
__global__ void gemm_wmma(const float* __restrict__ X, const float* __restrict__ W,
                          float* __restrict__ Hout) {
    const int lane = threadIdx.x & 31;
    const int wv   = threadIdx.x >> 5;
    const int row0 = blockIdx.x * 16;
    const int col0 = wv * 16;
    if (col0 >= FOUT) return;  // whole-wave early exit (partial last tile case)

    const int mrow = lane & 15;   // M index (A) / N index (B,D)
    const int kh   = lane >> 4;   // lane-half selects K sub-pair

    // A fragment source: lanes 0-15 -> K = {k, k+1}; lanes 16-31 -> K = {k+2, k+3}
    const float* ap = X + (row0 + mrow) * KD + kh * 2;
    const int  ccol = col0 + mrow;
    const bool cok  = (ccol < FOUT);

    v8f acc = {0.f, 0.f, 0.f, 0.f, 0.f, 0.f, 0.f, 0.f};

#pragma unroll
    for (int k = 0; k < KD; k += 4) {
        v2f a, b;
        a.x = ap[k];
        a.y = ap[k + 1];
        // B fragment: VGPR0 holds K = {k, k+2} across lane halves, VGPR1 holds {k+1, k+3}
        const int kb = k + kh * 2;
        b.x = cok ? W[kb * FOUT + ccol] : 0.0f;
        b.y = cok ? W[(kb + 1) * FOUT + ccol] : 0.0f;
        // D = A*B + C  (fp32, 16x16x4)
        acc = __builtin_amdgcn_wmma_f32_16x16x4_f32(false, a, false, b,
                                                    (short)0, acc, false, false);
    }

    // D layout: VGPR i -> M = i (lanes 0-15) / M = i+8 (lanes 16-31), N = lane%16
    const int rbase = row0 + kh * 8;
#pragma unroll
    for (int i = 0; i < 8; ++i) {
        if (cok) Hout[(rbase + i) * OST + ccol] = acc[i];
    }
}

// ---------------- aggregation ----------------

// Initialize accumulator with the self-loop term: out[i] = h[i] * dinv[i]^2
template <int F>
__global__ void self_loop_init(const float* __restrict__ h, const float* __restrict__ dinv,
                               float* __restrict__ out, int n) {
    int t = blockIdx.x * blockDim.x + threadIdx.x;
    if (t < n * F) {
        float d = dinv[t / F];
        out[t] = h[t] * d * d;
    }
}

// One thread per (edge, float-pair): out[dst] += h[src] * norm[e]
template <int F>
__global__ void edge_aggregate(const int* __restrict__ src, const int* __restrict__ dst,
                               const float* __restrict__ norm, const float* __restrict__ h,
                               float* out, int e) {
    const int P = F / 2;  // float pairs per edge
    int t  = blockIdx.x * blockDim.x + threadIdx.x;
    int ed = t / P;
    if (ed >= e) return;
    int j = (t - ed * P) * 2;
    int s = src[ed], d = dst[ed];
    float w = norm[ed];
    const float* hp = h + s * F + j;
    float* op = out + d * F + j;
    atomicAdd(op,     hp[0] * w);
    atomicAdd(op + 1, hp[1] * w);
}

// out = (relu?)(out + bias)
template <int F, bool RELU>
__global__ void bias_act(float* x, const float* __restrict__ b, int n) {
    int t = blockIdx.x * blockDim.x + threadIdx.x;
    if (t < n * F) {
        float v = x[t] + b[t % F];
        if (RELU) v = fmaxf(v, 0.0f);
        x[t] = v;
    }
}

// ---------------- driver ----------------

extern "C" void kernel_launch(void* const* d_in, const int* in_sizes, int n_in,
                              void* d_out, int out_size, void* d_ws, size_t ws_size,
                              hipStream_t stream) {
    (void)in_sizes; (void)n_in; (void)out_size; (void)ws_size;

    const float* x  = (const float*)d_in[0];
    const int*   ei = (const int*)d_in[1];   // [2, E]: row 0 = src, row 1 = dst
    const int*   src = ei;
    const int*   dst = ei + NEDGES;
    const float* W1 = (const float*)d_in[2];
    const float* b1 = (const float*)d_in[3];
    const float* W2 = (const float*)d_in[4];
    const float* b2 = (const float*)d_in[5];
    const float* W3 = (const float*)d_in[6];
    const float* b3 = (const float*)d_in[7];
    float* out = (float*)d_out;

    // Workspace layout (all fp32): dinv[N] | norm[E] | bufA[N*64] | bufB[N*64]
    float* dinv = (float*)d_ws;
    float* norm = dinv + NNODES;
    float* bufA = norm + NEDGES;
    float* bufB = bufA + NNODES * 64;

    const int BT = 256;
    const int gN   = (NNODES + BT - 1) / BT;
    const int gE   = (NEDGES + BT - 1) / BT;
    const int gN64 = (NNODES * 64 + BT - 1) / BT;
    const int gN40 = (NNODES * 40 + BT - 1) / BT;
    const int gE32 = (NEDGES / BT) * 32;            // E*32 / 256, exact
    const int gE20 = (NEDGES * 20 + BT - 1) / BT;   // fits in int: 32M
    const int gGemm = NNODES / 16;                  // 6250, exact

    // Shared normalization (computed once, reused by all 3 layers)
    init_deg   <<<gN, BT, 0, stream>>>(dinv, NNODES);
    count_deg  <<<gE, BT, 0, stream>>>(dst, dinv, NEDGES);
    deg_to_dinv<<<gN, BT, 0, stream>>>(dinv, NNODES);
    edge_norm  <<<gE, BT, 0, stream>>>(src, dst, dinv, norm, NEDGES);

    // Layer 1: h = relu(agg(x @ W1) + b1)
    gemm_wmma<64, 64, 64><<<gGemm, 128, 0, stream>>>(x, W1, bufA);
    self_loop_init<64>   <<<gN64, BT, 0, stream>>>(bufA, dinv, bufB, NNODES);
    edge_aggregate<64>   <<<gE32, BT, 0, stream>>>(src, dst, norm, bufA, bufB, NEDGES);
    bias_act<64, true>   <<<gN64, BT, 0, stream>>>(bufB, b1, NNODES);

    // Layer 2: h = relu(agg(h @ W2) + b2)
    gemm_wmma<64, 64, 64><<<gGemm, 128, 0, stream>>>(bufB, W2, bufA);
    self_loop_init<64>   <<<gN64, BT, 0, stream>>>(bufA, dinv, bufB, NNODES);
    edge_aggregate<64>   <<<gE32, BT, 0, stream>>>(src, dst, norm, bufA, bufB, NEDGES);
    bias_act<64, true>   <<<gN64, BT, 0, stream>>>(bufB, b2, NNODES);

    // Layer 3: out = agg(h @ W3) + b3   (C = 40, no relu)
    gemm_wmma<64, 40, 40><<<gGemm, 128, 0, stream>>>(bufB, W3, bufA);
    self_loop_init<40>   <<<gN40, BT, 0, stream>>>(bufA, dinv, out, NNODES);
    edge_aggregate<40>   <<<gE20, BT, 0, stream>>>(src, dst, norm, bufA, out, NEDGES);
    bias_act<40, false>  <<<gN40, BT, 0, stream>>>(out, b3, NNODES);
}